// IHSWrapper_29188597743946
// MI455X (gfx1250) — compile-verified
//
#include <hip/hip_runtime.h>
#include <hip/hip_bf16.h>

// Horn-Schunck iterations for MI455X (gfx1250).
//
// smooth(U) = (1/3) * (A*U*A - U), A = tridiag(0.5, 1.0, 0.5) 256x256 (zero-pad
// SAME == band truncation, exact). Vertical pass V = A*U and horizontal pass
// T = A*V^T = (A U A)^T are done with fp32 WMMA (V_WMMA_F32_16X16X4_F32),
// chaining K in blocks of 4; halo couplings live in the A-operand constants
// (one nonzero 0.5 entry each -> a single cndmask). Interior A-operands depend
// only on the lane and are hoisted out of all loops.
// Per-pixel update is an affine 2x2 mix with precomputed coefficient planes.
//
// Working set ~16MB -> L2 resident (192MB L2); 100 dependent Jacobi steps are
// launch/L2-latency bound, so we ping-pong 100 graph-captured launches.

#define IMG_H 256
#define IMG_W 256
#define IMG_HW (IMG_H * IMG_W)
#define NIMG 8

typedef __attribute__((ext_vector_type(2))) float v2f;
typedef __attribute__((ext_vector_type(4))) float v4f;
typedef __attribute__((ext_vector_type(8))) float v8f;

__device__ __forceinline__ int iclamp(int v, int lo, int hi) {
  return v < lo ? lo : (v > hi ? hi : v);
}

__device__ __forceinline__ float triv(int d) {
  return (d == 0) ? 1.0f : ((d == 1 || d == -1) ? 0.5f : 0.0f);
}

// Halo A-operands: exactly one nonzero coupling each.
// Left block (kb = m0-4): A[i=0][j=3] = 0.5  -> lane 16, .y component.
__device__ __forceinline__ v2f haloA_left(int lane, int m0) {
  v2f a;
  a.x = 0.0f;
  a.y = (lane == 16 && m0 > 0) ? 0.5f : 0.0f;
  return a;
}
// Right block (kb = m0+16): A[i=15][j=0] = 0.5 -> lane 15, .x component.
__device__ __forceinline__ v2f haloA_right(int lane, int m0) {
  v2f a;
  a.x = (lane == 15 && m0 + 16 < IMG_H) ? 0.5f : 0.0f;
  a.y = 0.0f;
  return a;
}

// Precompute affine coefficient planes and seed d_out with (u0, v0).
// u_new = c11*u_bar + c12*v_bar + cu ; v_new = c12*u_bar + c22*v_bar + cv
__global__ __launch_bounds__(256) void hs_pre(const float* __restrict__ x,
                                              const float* __restrict__ oe,
                                              float* __restrict__ coeff,
                                              float* __restrict__ out) {
  size_t i = (size_t)blockIdx.x * 256 + threadIdx.x;
  if (i >= (size_t)NIMG * IMG_HW) return;
  int n = (int)(i / IMG_HW);
  int p = (int)(i % IMG_HW);
  const size_t xb = (size_t)n * 3 * IMG_HW;
  float It = x[xb + p];
  float Ix = x[xb + IMG_HW + p];
  float Iy = x[xb + 2 * IMG_HW + p];
  float inv = 1.0f / (1.0f + Ix * Ix + Iy * Iy);  // LAMBDA_S^2 == 1
  const size_t P = (size_t)NIMG * IMG_HW;
  coeff[0 * P + i] = 1.0f - Ix * Ix * inv;
  coeff[1 * P + i] = -Ix * Iy * inv;
  coeff[2 * P + i] = 1.0f - Iy * Iy * inv;
  coeff[3 * P + i] = -Ix * It * inv;
  coeff[4 * P + i] = -Iy * It * inv;
  const size_t ob = (size_t)n * 2 * IMG_HW;
  out[ob + p] = oe[ob + p];                      // u0
  out[ob + IMG_HW + p] = oe[ob + IMG_HW + p];    // v0
}

// One Jacobi step for a 16-row x 256-col band. blockDim = 256 (8 wave32).
// grid = NIMG * 16 bands. Field f's plane: base + n*nstride + f*uvoff.
__global__ __launch_bounds__(256) void hs_iter(
    const float* __restrict__ src, int src_nstride, int src_uvoff,
    float* __restrict__ dst, int dst_nstride, int dst_uvoff,
    const float* __restrict__ coeff) {
  __shared__ float Vlds[2][16][IMG_W + 1];  // +1 pad: conflict-free col reads

  const int lane = threadIdx.x & 31;
  const int wave = threadIdx.x >> 5;
  const int n = blockIdx.x >> 4;
  const int r0 = (blockIdx.x & 15) << 4;
  const int j0 = (lane < 16) ? 0 : 2;

  const float* src_n = src + (size_t)n * src_nstride;

  // Interior tridiagonal A-operand blocks (k always in-range): lane-only.
  v2f aint[4];
  {
    const int i = lane & 15;
#pragma unroll
    for (int b = 0; b < 4; ++b) {
      aint[b].x = triv(i - (4 * b + j0));
      aint[b].y = triv(i - (4 * b + j0 + 1));
    }
  }

  // ---- Pass 1: vertical smoothing V = A * U (16 col-tiles x 2 fields) ----
#pragma unroll
  for (int s = 0; s < 4; ++s) {
    const int unit = wave + (s << 3);  // 8 waves cover 32 units
    const int f = unit >> 4;
    const int c0 = (unit & 15) << 4;
    const float* sp = src_n + f * src_uvoff;
    const int col = c0 + (lane & 15);
    v8f acc = {0.f, 0.f, 0.f, 0.f, 0.f, 0.f, 0.f, 0.f};
    {  // top halo block: only row r0-1 contributes (A kills the rest)
      const v2f a = haloA_left(lane, r0);
      v2f bm;
      bm.x = 0.0f;
      bm.y = sp[iclamp(r0 - 1, 0, IMG_H - 1) * IMG_W + col];
      acc = __builtin_amdgcn_wmma_f32_16x16x4_f32(false, a, false, bm,
                                                  (short)0, acc, false, false);
    }
#pragma unroll
    for (int b = 0; b < 4; ++b) {  // interior K blocks, rows r0..r0+15
      const int ra = r0 + 4 * b + j0;
      v2f bm;
      bm.x = sp[ra * IMG_W + col];
      bm.y = sp[(ra + 1) * IMG_W + col];
      acc = __builtin_amdgcn_wmma_f32_16x16x4_f32(
          false, aint[b], false, bm, (short)0, acc, false, false);
    }
    {  // bottom halo block: only row r0+16 contributes
      const v2f a = haloA_right(lane, r0);
      v2f bm;
      bm.x = sp[iclamp(r0 + 16, 0, IMG_H - 1) * IMG_W + col];
      bm.y = 0.0f;
      acc = __builtin_amdgcn_wmma_f32_16x16x4_f32(false, a, false, bm,
                                                  (short)0, acc, false, false);
    }
    const int lr = (lane < 16) ? 0 : 8;
#pragma unroll
    for (int j = 0; j < 8; ++j) Vlds[f][lr + j][col] = acc[j];
  }
  __syncthreads();

  // ---- Pass 2: T = A * V^T + (-U^T)  ==> (A U A - U)^T, then affine mix ----
#pragma unroll
  for (int s = 0; s < 2; ++s) {
    const int ct = wave + (s << 3);
    const int c0 = ct << 4;
    const int rloc = lane & 15;
    const int cb = c0 + ((lane < 16) ? 0 : 8);
    const int pix = (r0 + rloc) * IMG_W + cb;  // lane owns 8 consecutive px

    v8f accs[2];
#pragma unroll
    for (int f = 0; f < 2; ++f) {
      const float* sp = src_n + f * src_uvoff;
      const v4f ulo = *(const v4f*)(sp + pix);
      const v4f uhi = *(const v4f*)(sp + pix + 4);
      v8f c;
#pragma unroll
      for (int j = 0; j < 4; ++j) {
        c[j] = -ulo[j];
        c[j + 4] = -uhi[j];
      }
      accs[f] = c;
    }
    {  // left halo block: only column c0-1 contributes
      const v2f a = haloA_left(lane, c0);
      const int kc = iclamp(c0 - 4 + j0, 0, IMG_W - 2);
#pragma unroll
      for (int f = 0; f < 2; ++f) {
        const v2f bm = *(const v2f*)&Vlds[f][rloc][kc];
        accs[f] = __builtin_amdgcn_wmma_f32_16x16x4_f32(
            false, a, false, bm, (short)0, accs[f], false, false);
      }
    }
#pragma unroll
    for (int b = 0; b < 4; ++b) {  // interior K blocks, cols c0..c0+15
      const int kc = c0 + 4 * b + j0;
#pragma unroll
      for (int f = 0; f < 2; ++f) {
        const v2f bm = *(const v2f*)&Vlds[f][rloc][kc];
        accs[f] = __builtin_amdgcn_wmma_f32_16x16x4_f32(
            false, aint[b], false, bm, (short)0, accs[f], false, false);
      }
    }
    {  // right halo block: only column c0+16 contributes
      const v2f a = haloA_right(lane, c0);
      const int kc = iclamp(c0 + 16 + j0, 0, IMG_W - 2);
#pragma unroll
      for (int f = 0; f < 2; ++f) {
        const v2f bm = *(const v2f*)&Vlds[f][rloc][kc];
        accs[f] = __builtin_amdgcn_wmma_f32_16x16x4_f32(
            false, a, false, bm, (short)0, accs[f], false, false);
      }
    }

    // Pointwise affine update on this lane's 8 pixels.
    const size_t P = (size_t)NIMG * IMG_HW;
    const float* cbase = coeff + (size_t)n * IMG_HW + pix;
    v4f clo[5], chi[5];
#pragma unroll
    for (int q = 0; q < 5; ++q) {
      clo[q] = *(const v4f*)(cbase + (size_t)q * P);
      chi[q] = *(const v4f*)(cbase + (size_t)q * P + 4);
    }
    float un[8], vn[8];
#pragma unroll
    for (int j = 0; j < 8; ++j) {
      const float ub = accs[0][j] * (1.0f / 3.0f);
      const float vb = accs[1][j] * (1.0f / 3.0f);
      const int h = j & 3;
      const float a11 = (j < 4) ? clo[0][h] : chi[0][h];
      const float a12 = (j < 4) ? clo[1][h] : chi[1][h];
      const float a22 = (j < 4) ? clo[2][h] : chi[2][h];
      const float au = (j < 4) ? clo[3][h] : chi[3][h];
      const float av = (j < 4) ? clo[4][h] : chi[4][h];
      un[j] = a11 * ub + a12 * vb + au;
      vn[j] = a12 * ub + a22 * vb + av;
    }
    float* du = dst + (size_t)n * dst_nstride + pix;
    float* dv = du + dst_uvoff;
    v4f o;
    o[0] = un[0]; o[1] = un[1]; o[2] = un[2]; o[3] = un[3];
    *(v4f*)du = o;
    o[0] = un[4]; o[1] = un[5]; o[2] = un[6]; o[3] = un[7];
    *(v4f*)(du + 4) = o;
    o[0] = vn[0]; o[1] = vn[1]; o[2] = vn[2]; o[3] = vn[3];
    *(v4f*)dv = o;
    o[0] = vn[4]; o[1] = vn[5]; o[2] = vn[6]; o[3] = vn[7];
    *(v4f*)(dv + 4) = o;
  }
}

extern "C" void kernel_launch(void* const* d_in, const int* in_sizes, int n_in,
                              void* d_out, int out_size, void* d_ws,
                              size_t ws_size, hipStream_t stream) {
  (void)in_sizes; (void)n_in; (void)out_size; (void)ws_size;
  const float* x = (const float*)d_in[0];
  const float* oe = (const float*)d_in[1];
  float* out = (float*)d_out;

  const size_t P = (size_t)NIMG * IMG_HW;
  float* coeff = (float*)d_ws;  // 5 planes of P floats
  float* wsU = coeff + 5 * P;   // ping-pong planes: u at +0, v at +P

  const int total = NIMG * IMG_HW;
  hs_pre<<<(total + 255) / 256, 256, 0, stream>>>(x, oe, coeff, out);

  // 100 steps; start with (u0,v0) in d_out so the even iteration count ends
  // with the final state back in d_out (it=99 writes d_out).
  for (int it = 0; it < 100; ++it) {
    if ((it & 1) == 0) {
      hs_iter<<<NIMG * 16, 256, 0, stream>>>(out, 2 * IMG_HW, IMG_HW, wsU,
                                             IMG_HW, (int)P, coeff);
    } else {
      hs_iter<<<NIMG * 16, 256, 0, stream>>>(wsU, IMG_HW, (int)P, out,
                                             2 * IMG_HW, IMG_HW, coeff);
    }
  }
}